// MaxPool_74698071212039
// MI455X (gfx1250) — compile-verified
//
#include <hip/hip_runtime.h>
#include <hip/hip_bf16.h>

#define PATCH 7
#define THREADS 256
#define PLANE_GROUPS 16

// Kernel 1: compose the two gather levels ONCE, store transposed int32:
//   idxT[p * v_coarse + v] = vertices_to_prev_lvl[neihboring_patches[v*7 + p]]
// This removes ~2.3 GB of redundant index traffic (would otherwise be re-read
// per (b,c) plane) and makes the 7 index loads in the main kernel coalesced.
__global__ void build_idx_kernel(const int* __restrict__ vtp,
                                 const int* __restrict__ np,
                                 int* __restrict__ idxT,
                                 int v_coarse) {
    int v = blockIdx.x * blockDim.x + threadIdx.x;
    if (v >= v_coarse) return;
#pragma unroll
    for (int p = 0; p < PATCH; ++p) {
        int j = np[v * PATCH + p];      // coalesced-ish (28B/thread stride)
        int i = vtp[j];                 // random gather, 640KB table -> L2
        idxT[p * v_coarse + v] = i;     // fully coalesced store per p
    }
}

// Kernel 2: out[plane, v] = max_p x[plane, idxT[p, v]].
// One thread per coarse vertex v; 7 gather indices live in registers and are
// reused across planes/ PLANE_GROUPS strided planes per block. The plane id is
// block-uniform -> row base is an SGPR, gathers are saddr+voffset loads with
// voffsets hoisted out of the plane loop. Output uses a non-temporal store so
// the 84 MB write stream does not evict the x planes from L2.
__global__ void gather_max_kernel(const float* __restrict__ x,
                                  const int* __restrict__ idxT,
                                  float* __restrict__ out,
                                  int v_coarse, int v_fine, int planes) {
    int v = blockIdx.x * blockDim.x + threadIdx.x;
    if (v >= v_coarse) return;

    int i0 = idxT[0 * v_coarse + v];
    int i1 = idxT[1 * v_coarse + v];
    int i2 = idxT[2 * v_coarse + v];
    int i3 = idxT[3 * v_coarse + v];
    int i4 = idxT[4 * v_coarse + v];
    int i5 = idxT[5 * v_coarse + v];
    int i6 = idxT[6 * v_coarse + v];

    for (int plane = blockIdx.y; plane < planes; plane += PLANE_GROUPS) {
        const float* __restrict__ row = x + (size_t)plane * (size_t)v_fine;
        // Issue all 7 gathers up front (independent loads -> high MLP),
        // then reduce.
        float a0 = row[i0];
        float a1 = row[i1];
        float a2 = row[i2];
        float a3 = row[i3];
        float a4 = row[i4];
        float a5 = row[i5];
        float a6 = row[i6];
        float m = fmaxf(a0, a1);
        float n = fmaxf(a2, a3);
        float q = fmaxf(a4, a5);
        m = fmaxf(m, n);
        q = fmaxf(q, a6);
        m = fmaxf(m, q);
        __builtin_nontemporal_store(m, out + (size_t)plane * (size_t)v_coarse + v);
    }
}

// Fallback: fused variant (recomputes indices per plane-group) used only if
// the workspace cannot hold the 7*v_coarse int32 index table.
__global__ void gather_max_fused_kernel(const float* __restrict__ x,
                                        const int* __restrict__ vtp,
                                        const int* __restrict__ np,
                                        float* __restrict__ out,
                                        int v_coarse, int v_fine, int planes) {
    int v = blockIdx.x * blockDim.x + threadIdx.x;
    if (v >= v_coarse) return;

    int idx[PATCH];
#pragma unroll
    for (int p = 0; p < PATCH; ++p)
        idx[p] = vtp[np[v * PATCH + p]];

    for (int plane = blockIdx.y; plane < planes; plane += PLANE_GROUPS) {
        const float* __restrict__ row = x + (size_t)plane * (size_t)v_fine;
        float m = row[idx[0]];
#pragma unroll
        for (int p = 1; p < PATCH; ++p)
            m = fmaxf(m, row[idx[p]]);
        __builtin_nontemporal_store(m, out + (size_t)plane * (size_t)v_coarse + v);
    }
}

extern "C" void kernel_launch(void* const* d_in, const int* in_sizes, int n_in,
                              void* d_out, int out_size, void* d_ws, size_t ws_size,
                              hipStream_t stream) {
    const float* x   = (const float*)d_in[0];
    const int*   vtp = (const int*)d_in[1];
    const int*   np  = (const int*)d_in[2];
    float*       out = (float*)d_out;

    const int v_fine   = in_sizes[1];            // 163842
    const int v_coarse = in_sizes[2] / PATCH;    // 40962
    const int planes   = in_sizes[0] / v_fine;   // B*C = 512

    const int vblocks = (v_coarse + THREADS - 1) / THREADS;
    const size_t idx_bytes = (size_t)PATCH * (size_t)v_coarse * sizeof(int);

    if (ws_size >= idx_bytes) {
        int* idxT = (int*)d_ws;
        build_idx_kernel<<<dim3(vblocks), dim3(THREADS), 0, stream>>>(
            vtp, np, idxT, v_coarse);
        gather_max_kernel<<<dim3(vblocks, PLANE_GROUPS), dim3(THREADS), 0, stream>>>(
            x, idxT, out, v_coarse, v_fine, planes);
    } else {
        gather_max_fused_kernel<<<dim3(vblocks, PLANE_GROUPS), dim3(THREADS), 0, stream>>>(
            x, vtp, np, out, v_coarse, v_fine, planes);
    }
}